// RepresentationLoss_58007828299796
// MI455X (gfx1250) — compile-verified
//
#include <hip/hip_runtime.h>
#include <math.h>

// ---------------------------------------------------------------------------
// Problem constants (from the reference)
// ---------------------------------------------------------------------------
#define T_DIM 128
#define B_DIM 128
#define D_DIM 256
#define A_DIM 32
#define K_NEG 16
#define P_DIM 128
#define TAU   0.1f
#define EPSN  1e-12f
#define TM1B  ((T_DIM - 1) * B_DIM)   // 16256 anchor rows
#define TB    (T_DIM * B_DIM)         // 16384 total rows

typedef __attribute__((ext_vector_type(2))) float v2f;
typedef __attribute__((ext_vector_type(8))) float v8f;

__device__ __forceinline__ float wave_reduce_sum(float v) {
#pragma unroll
  for (int off = 16; off >= 1; off >>= 1) v += __shfl_xor(v, off, 32);
  return v;
}

// ---------------------------------------------------------------------------
// fp32 WMMA GEMM: C[M x NTOT] = act(A[M x KTOT] @ B[KTOT x NTOT] + bias)
// Block: 256 threads = 8 waves in 4(M) x 2(N); block tile 64 x 64.
// Each wave computes a 16x32 strip = two 16x16 WMMA tiles sharing one
// A fragment. B staged TRANSPOSED (n-major) so each lane's (k,k+1) pair is
// a single aligned ds_load_b64.
// ---------------------------------------------------------------------------
template <int KTOT, int NTOT, bool RELU>
__global__ void gemm_wmma(const float* __restrict__ Amat,
                          const float* __restrict__ Bmat,
                          const float* __restrict__ bias,
                          float* __restrict__ Cmat) {
  __shared__ float As[64][36];   // 64 rows x 32 k  (+pad; float4-aligned stores)
  __shared__ float Bt[64][34];   // 64 n   x 32 k  (transposed; even stride -> b64 frags)

  const int rowBase = blockIdx.x * 64;
  const int colBase = blockIdx.y * 64;
  const int tid  = threadIdx.x;
  const int wave = tid >> 5, lane = tid & 31;
  const int wm = wave >> 1, wn = wave & 1;     // 4(M) x 2(N) wave grid
  const int half = lane >> 4, l16 = lane & 15; // WMMA lane split

  v8f c0 = {}, c1 = {};

  for (int k0 = 0; k0 < KTOT; k0 += 32) {
    // --- stage A tile (64x32), float4 coalesced ---
    {
      const int r = tid >> 3, kc = (tid & 7) << 2;
#pragma unroll
      for (int s = 0; s < 2; ++s) {
        const float4 v =
            *(const float4*)&Amat[(size_t)(rowBase + r + s * 32) * KTOT + k0 + kc];
        *(float4*)&As[r + s * 32][kc] = v;
      }
    }
    // --- stage B tile (32k x 64n), coalesced float4 read, transposed store ---
    {
      const int kk = tid >> 4, n4 = (tid & 15) << 2;
#pragma unroll
      for (int s = 0; s < 2; ++s) {
        const float4 v =
            *(const float4*)&Bmat[(size_t)(k0 + kk + s * 16) * NTOT + colBase + n4];
        Bt[n4 + 0][kk + s * 16] = v.x;
        Bt[n4 + 1][kk + s * 16] = v.y;
        Bt[n4 + 2][kk + s * 16] = v.z;
        Bt[n4 + 3][kk + s * 16] = v.w;
      }
    }
    if (k0 + 32 < KTOT) {  // hint next K slab into cache (global_prefetch_b8)
      __builtin_prefetch(&Amat[(size_t)(rowBase + (tid >> 3)) * KTOT + k0 + 32], 0, 1);
      __builtin_prefetch(&Bmat[(size_t)(k0 + 32 + (tid >> 4)) * NTOT + colBase], 0, 1);
    }
    __syncthreads();

#pragma unroll
    for (int kk = 0; kk < 8; ++kk) {
      const int kb = kk * 4 + half * 2;  // even -> 8B-aligned b64 fragment loads
      v2f a, b0, b1;
      a  = *(const v2f*)&As[wm * 16 + l16][kb];
      b0 = *(const v2f*)&Bt[wn * 32 + l16][kb];
      b1 = *(const v2f*)&Bt[wn * 32 + 16 + l16][kb];
      c0 = __builtin_amdgcn_wmma_f32_16x16x4_f32(false, a, false, b0,
                                                 (short)0, c0, false, false);
      c1 = __builtin_amdgcn_wmma_f32_16x16x4_f32(false, a, false, b1,
                                                 (short)0, c1, false, false);
    }
    __syncthreads();
  }

#pragma unroll
  for (int s = 0; s < 2; ++s) {
    const v8f& c = s ? c1 : c0;
    const int col = colBase + wn * 32 + s * 16 + l16;
    const float bv = bias[col];
#pragma unroll
    for (int j = 0; j < 8; ++j) {
      const int row = rowBase + wm * 16 + half * 8 + j;  // C layout: VGPR j -> M = j / j+8
      float v = c[j] + bv;
      if (RELU) v = fmaxf(v, 0.0f);
      Cmat[(size_t)row * NTOT + col] = v;
    }
  }
}

// ---------------------------------------------------------------------------
// In-place l2 row normalization of P[TB x P_DIM]
// ---------------------------------------------------------------------------
__global__ void normalize_rows(float* __restrict__ P) {
  const int r = blockIdx.x * blockDim.x + threadIdx.x;
  if (r >= TB) return;
  float* row = P + (size_t)r * P_DIM;
  float ss = 0.0f;
#pragma unroll 4
  for (int i = 0; i < P_DIM; ++i) ss += row[i] * row[i];
  const float inv = 1.0f / fmaxf(sqrtf(ss), EPSN);
#pragma unroll 4
  for (int i = 0; i < P_DIM; ++i) row[i] *= inv;
}

// ---------------------------------------------------------------------------
// InfoNCE: one wave32 per (t,b) anchor; lanes hold 4 proj dims each.
// ---------------------------------------------------------------------------
__global__ void contrast_kernel(const float* __restrict__ Pn,
                                const unsigned char* __restrict__ mask,
                                const int* __restrict__ offsets,
                                const int* __restrict__ neg_idx,
                                float* __restrict__ acc) {
  const int gw = blockIdx.x * 4 + (threadIdx.x >> 5);
  if (gw >= TM1B) return;
  const int lane = threadIdx.x & 31;
  const int t = gw >> 7, b = gw & (B_DIM - 1);
  int tp = t + offsets[t];
  if (tp > T_DIM - 1) tp = T_DIM - 1;
  const int ra = t * B_DIM + b;

  const float4 av = *(const float4*)&Pn[(size_t)ra * P_DIM + lane * 4];

  float s[K_NEG + 1];
  {
    const float4 pv = *(const float4*)&Pn[(size_t)(tp * B_DIM + b) * P_DIM + lane * 4];
    float p = av.x * pv.x + av.y * pv.y + av.z * pv.z + av.w * pv.w;
    s[0] = wave_reduce_sum(p) * (1.0f / TAU);
  }
#pragma unroll
  for (int k = 0; k < K_NEG; ++k) {
    const int tn = neg_idx[(size_t)ra * K_NEG + k];
    const float4 pv = *(const float4*)&Pn[(size_t)(tn * B_DIM + b) * P_DIM + lane * 4];
    float p = av.x * pv.x + av.y * pv.y + av.z * pv.z + av.w * pv.w;
    s[k + 1] = wave_reduce_sum(p) * (1.0f / TAU);
  }

  float m = s[0];
#pragma unroll
  for (int i = 1; i <= K_NEG; ++i) m = fmaxf(m, s[i]);
  float sum = 0.0f;
#pragma unroll
  for (int i = 0; i <= K_NEG; ++i) sum += __expf(s[i] - m);
  const float per = m + __logf(sum) - s[0];

  const float valid = (mask[ra] && mask[tp * B_DIM + b]) ? 1.0f : 0.0f;
  if (lane == 0) {
    atomicAdd(&acc[0], per * valid);
    atomicAdd(&acc[1], valid);
  }
}

// ---------------------------------------------------------------------------
// Temporal consistency: ||z[t]-z[t+1]|| per (t,b); one wave per row.
// ---------------------------------------------------------------------------
__global__ void tc_kernel(const float* __restrict__ z,
                          const unsigned char* __restrict__ mask,
                          float* __restrict__ acc) {
  const int gw = blockIdx.x * 4 + (threadIdx.x >> 5);
  if (gw >= TM1B) return;
  const int lane = threadIdx.x & 31;
  const size_t b0 = (size_t)gw * D_DIM + lane * 8;
  const size_t b1 = (size_t)(gw + B_DIM) * D_DIM + lane * 8;
  float ss = 0.0f;
#pragma unroll
  for (int s = 0; s < 2; ++s) {
    const float4 x = *(const float4*)&z[b0 + s * 4];
    const float4 y = *(const float4*)&z[b1 + s * 4];
    const float dx = x.x - y.x, dy = x.y - y.y, dz = x.z - y.z, dw = x.w - y.w;
    ss += dx * dx + dy * dy + dz * dz + dw * dw;
  }
  ss = wave_reduce_sum(ss);
  if (lane == 0) {
    const float v2 = (mask[gw] && mask[gw + B_DIM]) ? 1.0f : 0.0f;
    atomicAdd(&acc[2], sqrtf(ss) * v2);
    atomicAdd(&acc[3], v2);
  }
}

// ---------------------------------------------------------------------------
// Fused prediction loss: pred = concat(z,a)[16256x288] @ Wd + bd, vs z[1:].
// Block tile 16 x 256 (full N) so per-row squared error completes in-block.
// 8 waves, each wave owns two 16x16 column tiles (covers N=256).
// ---------------------------------------------------------------------------
__global__ void pred_gemm_loss(const float* __restrict__ z,
                               const float* __restrict__ aact,
                               const unsigned char* __restrict__ mask,
                               const float* __restrict__ Wd,
                               const float* __restrict__ bd,
                               float* __restrict__ acc) {
  __shared__ float As[16][36];    // 16 rows x 32 k
  __shared__ float Bs[32][260];   // 32 k x 256 n
  __shared__ float rowsq[16];

  const int rowBase = blockIdx.x * 16;   // 1016 blocks x 16 = 16256
  const int tid  = threadIdx.x;
  const int wave = tid >> 5, lane = tid & 31;
  const int half = lane >> 4, l16 = lane & 15;

  if (tid < 16) rowsq[tid] = 0.0f;

  v8f c0 = {}, c1 = {};
  for (int k0 = 0; k0 < D_DIM + A_DIM; k0 += 32) {
    // --- stage A (16x32): X[row][k] = (k<256) ? z[row][k] : a[row][k-256] ---
    {
      const int r = tid >> 4, kc = (tid & 15) << 1;
      const int row = rowBase + r;
#pragma unroll
      for (int s = 0; s < 2; ++s) {
        const int k = k0 + kc + s;
        As[r][kc + s] = (k < D_DIM) ? z[(size_t)row * D_DIM + k]
                                    : aact[(size_t)row * A_DIM + (k - D_DIM)];
      }
    }
    // --- stage B (32x256) from Wd, float4 coalesced ---
    {
      const int n = (tid & 63) << 2;
#pragma unroll
      for (int kk = tid >> 6; kk < 32; kk += 4) {
        const float4 v = *(const float4*)&Wd[(size_t)(k0 + kk) * D_DIM + n];
        *(float4*)&Bs[kk][n] = v;
      }
    }
    __syncthreads();

#pragma unroll
    for (int kk = 0; kk < 8; ++kk) {
      const int kb = kk * 4 + half * 2;
      v2f a, b0, b1v;
      a.x = As[l16][kb];
      a.y = As[l16][kb + 1];
      b0.x  = Bs[kb][wave * 32 + l16];
      b0.y  = Bs[kb + 1][wave * 32 + l16];
      b1v.x = Bs[kb][wave * 32 + 16 + l16];
      b1v.y = Bs[kb + 1][wave * 32 + 16 + l16];
      c0 = __builtin_amdgcn_wmma_f32_16x16x4_f32(false, a, false, b0,
                                                 (short)0, c0, false, false);
      c1 = __builtin_amdgcn_wmma_f32_16x16x4_f32(false, a, false, b1v,
                                                 (short)0, c1, false, false);
    }
    __syncthreads();
  }

  // epilogue: squared diff against target z[row + B] (= z[t+1, b])
#pragma unroll
  for (int s = 0; s < 2; ++s) {
    const v8f& c = s ? c1 : c0;
    const int col = wave * 32 + s * 16 + l16;
    const float bv = bd[col];
#pragma unroll
    for (int j = 0; j < 8; ++j) {
      const int rloc = half * 8 + j;
      const int row = rowBase + rloc;
      const float d = c[j] + bv - z[(size_t)(row + B_DIM) * D_DIM + col];
      atomicAdd(&rowsq[rloc], d * d);
    }
  }
  __syncthreads();

  if (tid < 16) {
    const int row = rowBase + tid;
    const int t = row >> 7, b = row & (B_DIM - 1);
    const float v2 = (mask[t * B_DIM + b] && mask[(t + 1) * B_DIM + b]) ? 1.0f : 0.0f;
    atomicAdd(&acc[4], sqrtf(rowsq[tid]) * v2);
    atomicAdd(&acc[5], v2);
  }
}

__global__ void zero_acc(float* __restrict__ acc) {
  if (threadIdx.x < 8) acc[threadIdx.x] = 0.0f;
}

__global__ void finalize(const float* __restrict__ acc, float* __restrict__ out) {
  const float contrast = acc[0] / fmaxf(acc[1], 1.0f);
  const float tc       = acc[2] / fmaxf(acc[3], 1.0f);
  const float pr       = acc[4] / fmaxf(acc[5], 1.0f);
  out[0] = contrast;
  out[1] = tc;
  out[2] = pr;
  out[3] = contrast + tc + pr;  // LC = LT = LP = 1
}

// ---------------------------------------------------------------------------
// Launch
// ---------------------------------------------------------------------------
extern "C" void kernel_launch(void* const* d_in, const int* in_sizes, int n_in,
                              void* d_out, int out_size, void* d_ws, size_t ws_size,
                              hipStream_t stream) {
  const float* z  = (const float*)d_in[0];
  const float* a  = (const float*)d_in[1];
  const unsigned char* mask = (const unsigned char*)d_in[2];
  const int* offsets = (const int*)d_in[3];
  const int* neg_idx = (const int*)d_in[4];
  const float* W1 = (const float*)d_in[5];
  const float* b1 = (const float*)d_in[6];
  const float* W2 = (const float*)d_in[7];
  const float* b2 = (const float*)d_in[8];
  const float* Wd = (const float*)d_in[9];
  const float* bd = (const float*)d_in[10];
  float* out = (float*)d_out;

  char* ws = (char*)d_ws;
  float* H   = (float*)ws;                                         // TB x D
  float* P   = (float*)(ws + (size_t)TB * D_DIM * sizeof(float));  // TB x P
  float* acc = (float*)(ws + (size_t)TB * D_DIM * sizeof(float)
                           + (size_t)TB * P_DIM * sizeof(float));  // 8 floats

  zero_acc<<<1, 32, 0, stream>>>(acc);

  // H = relu(Z @ W1 + b1)        [16384 x 256]
  gemm_wmma<D_DIM, D_DIM, true>
      <<<dim3(TB / 64, D_DIM / 64), 256, 0, stream>>>(z, W1, b1, H);

  // P = H @ W2 + b2              [16384 x 128]  (then l2-normalize rows)
  gemm_wmma<D_DIM, P_DIM, false>
      <<<dim3(TB / 64, P_DIM / 64), 256, 0, stream>>>(H, W2, b2, P);
  normalize_rows<<<TB / 256, 256, 0, stream>>>(P);

  // InfoNCE over gathered normalized projections
  contrast_kernel<<<TM1B / 4, 128, 0, stream>>>(P, mask, offsets, neg_idx, acc);

  // Temporal consistency
  tc_kernel<<<TM1B / 4, 128, 0, stream>>>(z, mask, acc);

  // Fused prediction GEMM + loss
  pred_gemm_loss<<<TM1B / 16, 256, 0, stream>>>(z, a, mask, Wd, bd, acc);

  finalize<<<1, 1, 0, stream>>>(acc, out);
}